// AuxiliaryLoss_49520972922932
// MI455X (gfx1250) — compile-verified
//
#include <hip/hip_runtime.h>
#include <hip/hip_bf16.h>

// ---------------------------------------------------------------------------
// AuxiliaryLoss (FAPE + torsion) for MI455X / gfx1250.
//
// FAPE pairwise term is re-cast as three N x N f32 GEMMs with K=8:
//   diff_o(i,j) = u_o(i) . v(j),  u_o in R^8 (row-side), v in R^8 (col-side)
// computed with V_WMMA_F32_16X16X4_F32 (two K=4 WMMAs per axis per tile),
// fused with sqrt/clamp/mean reduction. All data is L2-resident (~4 MB).
// sqrt/rcp use raw TRANS ops (args are >= 1e-4, no denormal fixup needed).
// Sum-of-squares is expressed as whole-v8f vector math so the compiler emits
// straight v_pk_mul/v_pk_fma on adjacent components without register shuffles.
// ---------------------------------------------------------------------------

typedef __attribute__((ext_vector_type(2))) float v2f;
typedef __attribute__((ext_vector_type(8))) float v8f;

#define LN 8       // layers
#define BN 2       // batch
#define NN 2048    // residues
#define JT 16      // j-tiles (16 cols each) per wave

static __device__ inline v8f wmma4(v2f a, v2f b, v8f c) {
  // 8 args: (neg_a, A, neg_b, B, c_mod, C, reuse_a, reuse_b)
  return __builtin_amdgcn_wmma_f32_16x16x4_f32(
      false, a, false, b, (short)0, c, false, false);
}

// ---------------------------------------------------------------------------
// Kernel 0: zero the per-batch accumulators in workspace.
// ---------------------------------------------------------------------------
__global__ void zero_accum_kernel(float* __restrict__ accum) {
  if (threadIdx.x < BN) accum[threadIdx.x] = 0.0f;
}

// ---------------------------------------------------------------------------
// Kernel 1: precompute A-side (u_o) and B-side (v) 8-vectors.
//   Apre[((lb*3+o)*N + i)*8 + k],  Bpre[(lb*N + j)*8 + k]
// One thread per (l,b,n).
// ---------------------------------------------------------------------------
__global__ void precompute_kernel(const float* __restrict__ Rp_all,   // [L,B,N,3,3]
                                  const float* __restrict__ tp_all,   // [L,B,N,3]
                                  const float* __restrict__ Rt_all,   // [B,N,3,3]
                                  const float* __restrict__ tt_all,   // [B,N,3]
                                  float* __restrict__ Apre,
                                  float* __restrict__ Bpre) {
  int t = blockIdx.x * blockDim.x + threadIdx.x;
  if (t >= LN * BN * NN) return;
  int n  = t % NN;
  int lb = t / NN;
  int b  = lb % BN;

  const float* Rp = Rp_all + (size_t)t * 9;
  const float* tp = tp_all + (size_t)t * 3;
  const float* Rt = Rt_all + ((size_t)b * NN + n) * 9;
  const float* tt = tt_all + ((size_t)b * NN + n) * 3;

  float tp0 = tp[0], tp1 = tp[1], tp2 = tp[2];
  float tt0 = tt[0], tt1 = tt[1], tt2 = tt[2];

#pragma unroll
  for (int o = 0; o < 3; ++o) {
    // Rp_inv[o,p] = Rp[p,o]  (transpose)
    float rp0 = Rp[0 * 3 + o], rp1 = Rp[1 * 3 + o], rp2 = Rp[2 * 3 + o];
    float rt0 = Rt[0 * 3 + o], rt1 = Rt[1 * 3 + o], rt2 = Rt[2 * 3 + o];
    float tp_inv = -(rp0 * tp0 + rp1 * tp1 + rp2 * tp2);
    float tt_inv = -(rt0 * tt0 + rt1 * tt1 + rt2 * tt2);
    float* A = Apre + (((size_t)(lb * 3 + o)) * NN + n) * 8;
    A[0] = rp0;  A[1] = rp1;  A[2] = rp2;  A[3] = tp_inv;
    A[4] = -rt0; A[5] = -rt1; A[6] = -rt2; A[7] = -tt_inv;
  }
  float* Bv = Bpre + ((size_t)lb * NN + n) * 8;
  Bv[0] = tp0; Bv[1] = tp1; Bv[2] = tp2; Bv[3] = 1.0f;
  Bv[4] = tt0; Bv[5] = tt1; Bv[6] = tt2; Bv[7] = 1.0f;
}

// ---------------------------------------------------------------------------
// Kernel 2: FAPE via WMMA. One wave per (lb, i_tile, j_chunk); the wave owns a
// 16-row x (16*JT)-col strip of the N x N distance matrix. Grid is exact so
// every wave has EXEC = all-ones (WMMA requirement).
// ---------------------------------------------------------------------------
__global__ void fape_wmma_kernel(const float* __restrict__ Apre,
                                 const float* __restrict__ Bpre,
                                 float* __restrict__ accum) {
  const int lane   = threadIdx.x & 31;
  const int wave   = (blockIdx.x * blockDim.x + threadIdx.x) >> 5;
  const int nJC    = NN / (16 * JT);          // 8 j-chunks
  const int nIT    = NN / 16;                 // 128 i-tiles
  const int jc     = wave % nJC;
  const int it     = (wave / nJC) % nIT;
  const int lb     = wave / (nJC * nIT);      // 0..L*B-1
  const int b      = lb % BN;

  const int laneM  = lane & 15;               // row/col within tile
  const int laneHi = lane >> 4;               // K sub-pair select (ISA layout)
  const int i0     = it * 16;
  const int j0     = jc * 16 * JT;

  // A fragments: a[o][kb] covers K = 4*kb .. 4*kb+3 of u_o.
  // Lane layout (ISA 7.12.2, 32-bit A 16x4): lanes 0-15 -> K={0,1},
  // lanes 16-31 -> K={2,3}; row M = lane % 16. 8B-aligned float2 loads.
  v2f a[3][2];
#pragma unroll
  for (int o = 0; o < 3; ++o)
#pragma unroll
    for (int kb = 0; kb < 2; ++kb)
      a[o][kb] = *(const v2f*)(Apre +
          (((size_t)(lb * 3 + o)) * NN + i0 + laneM) * 8 + 4 * kb + 2 * laneHi);

  float sum = 0.0f;
  for (int jt = 0; jt < JT; ++jt) {
    const int j = j0 + jt * 16 + laneM;
    const float* bp = Bpre + ((size_t)lb * NN + j) * 8;
    v2f b0 = *(const v2f*)(bp + 0 + 2 * laneHi);   // K = 0..3
    v2f b1 = *(const v2f*)(bp + 4 + 2 * laneHi);   // K = 4..7

    v8f d0 = {}, d1 = {}, d2 = {};
    d0 = wmma4(a[0][0], b0, d0);  d0 = wmma4(a[0][1], b1, d0);
    d1 = wmma4(a[1][0], b0, d1);  d1 = wmma4(a[1][1], b1, d1);
    d2 = wmma4(a[2][0], b0, d2);  d2 = wmma4(a[2][1], b1, d2);

    // Whole-vector sum of squares: lowers to v_pk_mul/v_pk_fma/v_pk_add on
    // adjacent components of each accumulator (no cross-register shuffles).
    v8f dd = d0 * d0 + d1 * d1 + d2 * d2 + 1e-4f;

#pragma unroll
    for (int s = 0; s < 8; ++s) {
      // dd[s] >= 1e-4: raw v_sqrt_f32 (TRANS, co-issues with VALU/WMMA),
      // no IEEE denormal-guard / ulp-fixup expansion needed.
      sum += fminf(__builtin_amdgcn_sqrtf(dd[s]), 10.0f);
    }
  }

  // Wave32 reduction, then one atomic per wave with all scaling folded in:
  // fape contribution = sum / (N*N) / Z / L
#pragma unroll
  for (int off = 16; off > 0; off >>= 1)
    sum += __shfl_down(sum, off, 32);
  if (lane == 0) {
    const float scale = 1.0f / ((float)NN * (float)NN * 10.0f * (float)LN);
    atomicAdd(&accum[b], sum * scale);
  }
}

// ---------------------------------------------------------------------------
// Kernel 3: torsion angle loss. One thread per (l,b,n); 7 torsion pairs each.
// ---------------------------------------------------------------------------
__global__ void torsion_kernel(const float* __restrict__ tor,   // [L,B,N,7,2]
                               const float* __restrict__ ttrue, // [B,N,7,2]
                               const float* __restrict__ talt,  // [B,N,7,2]
                               float* __restrict__ accum) {
  int t = blockIdx.x * blockDim.x + threadIdx.x;
  if (t >= LN * BN * NN) return;
  int n  = t % NN;
  int lb = t / NN;
  int b  = lb % BN;

  const float* p  = tor   + (size_t)t * 14;
  const float* tv = ttrue + ((size_t)b * NN + n) * 14;
  const float* ta = talt  + ((size_t)b * NN + n) * 14;

  float st = 0.0f, sa = 0.0f;
#pragma unroll
  for (int k = 0; k < 7; ++k) {
    float px = p[2 * k], py = p[2 * k + 1];
    float n2  = px * px + py * py + 1e-8f;
    float nrm = __builtin_amdgcn_sqrtf(n2);      // arg >= 1e-8, no guard needed
    float inv = __builtin_amdgcn_rcpf(nrm);      // v_rcp_f32
    float pnx = px * inv, pny = py * inv;
    float dx  = tv[2 * k] - pnx,     dy  = tv[2 * k + 1] - pny;
    float dxa = ta[2 * k] - pnx,     dya = ta[2 * k + 1] - pny;
    float dT  = __builtin_amdgcn_sqrtf(dx * dx + dy * dy + 1e-8f);
    float dA  = __builtin_amdgcn_sqrtf(dxa * dxa + dya * dya + 1e-8f);
    st += fminf(dT, dA);
    sa += fabsf(nrm - 1.0f);
  }
  // per-residue loss, scaled by 1/(N*L) for the mean over N and layers
  float v = (st * (1.0f / 7.0f) + 0.02f * sa * (1.0f / 7.0f)) *
            (1.0f / ((float)NN * (float)LN));

  // lanes of one wave share b (N % 32 == 0): wave-reduce then one atomic
#pragma unroll
  for (int off = 16; off > 0; off >>= 1)
    v += __shfl_down(v, off, 32);
  if ((threadIdx.x & 31) == 0) atomicAdd(&accum[b], v);
}

// ---------------------------------------------------------------------------
// Kernel 4: write accumulators to d_out.
// ---------------------------------------------------------------------------
__global__ void finalize_kernel(const float* __restrict__ accum,
                                float* __restrict__ out) {
  if (threadIdx.x < BN) out[threadIdx.x] = accum[threadIdx.x];
}

// ---------------------------------------------------------------------------
extern "C" void kernel_launch(void* const* d_in, const int* in_sizes, int n_in,
                              void* d_out, int out_size, void* d_ws, size_t ws_size,
                              hipStream_t stream) {
  const float* traj_rot = (const float*)d_in[0];  // [L,B,N,3,3]
  const float* traj_trn = (const float*)d_in[1];  // [L,B,N,3]
  const float* traj_tor = (const float*)d_in[2];  // [L,B,N,7,2]
  const float* true_rot = (const float*)d_in[3];  // [B,N,3,3]
  const float* true_trn = (const float*)d_in[4];  // [B,N,3]
  const float* true_tor = (const float*)d_in[5];  // [B,N,7,2]
  const float* true_tora= (const float*)d_in[6];  // [B,N,7,2]
  float* out = (float*)d_out;

  // Workspace layout (floats): [accum:16][Apre: L*B*3*N*8][Bpre: L*B*N*8]
  float* ws    = (float*)d_ws;
  float* accum = ws;
  float* Apre  = ws + 16;
  float* Bpre  = Apre + (size_t)LN * BN * 3 * NN * 8;

  zero_accum_kernel<<<1, 64, 0, stream>>>(accum);

  {
    int total = LN * BN * NN;                     // 32768 threads
    precompute_kernel<<<(total + 255) / 256, 256, 0, stream>>>(
        traj_rot, traj_trn, true_rot, true_trn, Apre, Bpre);
  }

  {
    // waves = L*B * (N/16) * (N/(16*JT)) = 16 * 128 * 8 = 16384
    int waves  = LN * BN * (NN / 16) * (NN / (16 * JT));
    int blocks = waves * 32 / 256;                // exact: 2048 blocks
    fape_wmma_kernel<<<blocks, 256, 0, stream>>>(Apre, Bpre, accum);
  }

  {
    int total = LN * BN * NN;
    torsion_kernel<<<(total + 255) / 256, 256, 0, stream>>>(
        traj_tor, true_tor, true_tora, accum);
  }

  finalize_kernel<<<1, 64, 0, stream>>>(accum, out);
}